// VSmodel_69295002353729
// MI455X (gfx1250) — compile-verified
//
#include <hip/hip_runtime.h>
#include <hip/hip_bf16.h>

#define PP      512
#define DD      128
#define NMOLS   262144
#define NRAND   10000
#define NSEG    32
#define TOPK    100000

typedef __attribute__((ext_vector_type(2))) float    v2f;
typedef __attribute__((ext_vector_type(8))) float    v8f;
typedef __attribute__((ext_vector_type(4))) unsigned v4u;
typedef __attribute__((ext_vector_type(4))) int      v4i;
typedef __attribute__((ext_vector_type(8))) int      v8i;

// Monotonic float <-> uint mapping (ascending order preserved).
__device__ __forceinline__ unsigned ordf(float f) {
    unsigned b = __float_as_uint(f);
    return (b & 0x80000000u) ? ~b : (b | 0x80000000u);
}
__device__ __forceinline__ float unordf(unsigned u) {
    unsigned b = (u & 0x80000000u) ? (u ^ 0x80000000u) : ~u;
    return __uint_as_float(b);
}

// 16-byte global -> LDS copy. Prefers the gfx1250 async-to-LDS path
// (tracked with ASYNCcnt); falls back to a VGPR round-trip copy.
__device__ __forceinline__ void async_copy16(const void* g, void* l) {
#if __has_builtin(__builtin_amdgcn_global_load_async_to_lds_b128)
    __builtin_amdgcn_global_load_async_to_lds_b128(
        (__attribute__((address_space(1))) v4i*)(unsigned long long)g,
        (__attribute__((address_space(3))) v4i*)(unsigned)(unsigned long long)l,
        0, 0);
#else
    *(float4*)l = *(const float4*)g;
#endif
}

__device__ __forceinline__ void wait_async0() {
#if __has_builtin(__builtin_amdgcn_s_wait_asynccnt)
    __builtin_amdgcn_s_wait_asynccnt(0);
#endif
}

// ---------------------------------------------------------------------------
// Kernel 1: per-row median + MAD via exact 4-level radix select in LDS.
// One block (256 threads) per row of input_array.
// ---------------------------------------------------------------------------
__device__ unsigned radix_select_k(const float* samp, int n, int k, float sub,
                                   int absmode, unsigned* hist, int tid) {
    unsigned prefix = 0;
    int kk = k;
    for (int level = 3; level >= 0; --level) {
        for (int b = tid; b < 256; b += 256) hist[b] = 0u;
        __syncthreads();
        unsigned himask = (level == 3) ? 0u : (0xFFFFFFFFu << ((level + 1) * 8));
        for (int i = tid; i < n; i += 256) {
            float v = samp[i];
            if (absmode) v = fabsf(v - sub);
            unsigned u = ordf(v);
            if ((u & himask) == prefix)
                atomicAdd(&hist[(u >> (level * 8)) & 255u], 1u);
        }
        __syncthreads();
        unsigned cum = 0; int bsel = 255;
        for (int b = 0; b < 256; ++b) {
            unsigned c = hist[b];
            if ((unsigned)kk < cum + c) { bsel = b; break; }
            cum += c;
        }
        kk -= (int)cum;
        prefix |= ((unsigned)bsel) << (level * 8);
        __syncthreads();
    }
    return prefix;  // exact ascending key of k-th smallest
}

__global__ __launch_bounds__(256) void vs_stats_kernel(
    const float* __restrict__ input, const float* __restrict__ rnd,
    float* __restrict__ med_out, float* __restrict__ scale_out) {
    __shared__ float srow[DD];
    __shared__ float samp[NRAND];
    __shared__ unsigned hist[256];
    int p = blockIdx.x, tid = threadIdx.x;

    for (int i = tid; i < DD; i += 256) srow[i] = input[p * DD + i];
    __syncthreads();

    for (int j = tid; j < NRAND; j += 256) {
        float acc = 0.f;
        #pragma unroll 8
        for (int kk = 0; kk < DD; ++kk)
            acc = fmaf(srow[kk], rnd[(size_t)kk * NRAND + j], acc);
        samp[j] = acc;
    }
    __syncthreads();

    // jnp.median on 10000 elems = mean of order stats 4999 & 5000 (0-indexed)
    unsigned u0 = radix_select_k(samp, NRAND, 4999, 0.f, 0, hist, tid);
    unsigned u1 = radix_select_k(samp, NRAND, 5000, 0.f, 0, hist, tid);
    float med = 0.5f * (unordf(u0) + unordf(u1));
    unsigned m0 = radix_select_k(samp, NRAND, 4999, med, 1, hist, tid);
    unsigned m1 = radix_select_k(samp, NRAND, 5000, med, 1, hist, tid);
    float mad = 0.5f * (unordf(m0) + unordf(m1));
    if (tid == 0) {
        med_out[p] = med;
        scale_out[p] = 0.6745f / (mad + 1e-6f);
    }
}

// ---------------------------------------------------------------------------
// Kernel 2: fused f32 WMMA GEMM + z-score + segment-max + key packing.
// Block = 256 threads (8 waves). Block owns a 128-column stripe; wave w owns
// 16 columns. A (512x128 f32 = 256 KB) staged into LDS by one TDM descriptor
// (TENSORcnt) -- or async b128 copies -- and reused by all 8 waves. B stripe
// (128x16 per wave) lives in 64 VGPRs/lane. Emits u64 keys:
// (~asc(z_max) << 32) | col  => ascending sort == top_k order w/ jax ties.
// ---------------------------------------------------------------------------
#define K2_SMEM (262144 + 2048 + 2048 + 2048 + 16384)

__global__ __launch_bounds__(256) void vs_fused_gemm_segmax(
    const float* __restrict__ A, const float* __restrict__ B,
    const float* __restrict__ med, const float* __restrict__ scale,
    const int* __restrict__ pidx, unsigned long long* __restrict__ keys) {
    extern __shared__ char smem[];
    float*    Alds = (float*)smem;                     // 512*128 f32
    float*    medl = (float*)(smem + 262144);          // 512 f32
    float*    scl  = (float*)(smem + 264192);          // 512 f32
    int*      segl = (int*)  (smem + 266240);          // 512 i32
    unsigned* smax = (unsigned*)(smem + 268288);       // 32 seg x 128 col

    int tid = threadIdx.x;

    // ---- Stage A (256 KB) global -> LDS ---------------------------------
#if __has_builtin(__builtin_amdgcn_tensor_load_to_lds)
    if (tid < 32) {  // one TDM descriptor issued by wave 0 (EXEC ignored)
        unsigned long long ga = (unsigned long long)(const void*)A;
        unsigned ldsOff = (unsigned)(unsigned long long)(void*)Alds;
        v4u g0;
        g0.x = 1u;                                        // count=1 user D#
        g0.y = ldsOff;                                    // lds_addr
        g0.z = (unsigned)ga;                              // global_addr[31:0]
        g0.w = (unsigned)((ga >> 32) & 0x01FFFFFFull)     // global_addr[56:32]
               | 0x80000000u;                             // type=2 ("image")
        // 2-D tensor: dim0=128 elems (one row), dim1=512 rows, 4B elements,
        // tile == whole tensor, row stride = 128 elements.
        v8i g1;
        g1[0] = 0x00020000;          // data_size=2 (4 bytes), mask=0, flags=0
        g1[1] = (int)(128u << 16);   // tensor_dim0[15:0] = 128
        g1[2] = (int)(512u << 16);   // tensor_dim1[15:0] = 512
        g1[3] = (int)(128u << 16);   // tile_dim0 = 128
        g1[4] = 0x00000200;          // tile_dim1 = 512 (tile_dim2 = 0)
        g1[5] = 128;                 // tensor_dim0_stride = 128
        g1[6] = 0;
        g1[7] = 0;                   // tensor_dim1_stride unused (2-D)
        v4i zz = {0, 0, 0, 0};
        v8i z8 = {0, 0, 0, 0, 0, 0, 0, 0};
        // 6-arg flavor (upstream clang-23 / therock headers).
        __builtin_amdgcn_tensor_load_to_lds(g0, g1, zz, zz, z8, 0);
    }
#else
    {
        const float4* A4 = (const float4*)A;
        float4* L4 = (float4*)Alds;
        for (int i = tid; i < (PP * DD / 4); i += 256)
            async_copy16(A4 + i, L4 + i);
    }
#endif

    // ---- Stage med/scale/segment tables via async-to-LDS (ASYNCcnt) -----
    for (int i = tid; i < PP / 4; i += 256) {
        async_copy16(((const float4*)med) + i, ((float4*)medl) + i);
        async_copy16(((const float4*)scale) + i, ((float4*)scl) + i);
        async_copy16(((const int4*)pidx) + i, ((int4*)segl) + i);
    }
    unsigned NEGINF = ordf(-__builtin_inff());
    for (int i = tid; i < NSEG * 128; i += 256) smax[i] = NEGINF;

    wait_async0();
#if __has_builtin(__builtin_amdgcn_tensor_load_to_lds)
    if (tid < 32) __builtin_amdgcn_s_wait_tensorcnt(0);
#endif
    __syncthreads();

    int wave = tid >> 5, lane = tid & 31;
    int colBase = blockIdx.x * 128 + wave * 16;
    int cl = colBase + (lane & 15);
    int khalf = (lane >> 4) * 2;   // f32 A/B frag: vgpr0={K+0,K+2}, vgpr1={K+1,K+3}

    // Full-K B stripe in registers: 32 fragments (4x16 f32 each).
    v2f bfrag[32];
    #pragma unroll
    for (int kc = 0; kc < 32; ++kc) {
        int kk = kc * 4 + khalf;
        bfrag[kc].x = B[(size_t)kk * NMOLS + cl];
        bfrag[kc].y = B[(size_t)(kk + 1) * NMOLS + cl];
        // Pull the next column stripe toward L2 while we stream this one.
        __builtin_prefetch(&B[(size_t)kk * NMOLS + cl + 128], 0, 1);
    }

    int m = lane & 15;
    for (int r = 0; r < 32; ++r) {   // 32 row tiles of 16 cover P=512
        v8f acc = {0.f, 0.f, 0.f, 0.f, 0.f, 0.f, 0.f, 0.f};
        const float* arow = &Alds[(r * 16 + m) * DD + khalf];
        #pragma unroll
        for (int kc = 0; kc < 32; ++kc) {
            v2f af;
            af.x = arow[kc * 4];
            af.y = arow[kc * 4 + 1];
            // D = A(16x4 f32) * B(4x16 f32) + C(16x16 f32)
            acc = __builtin_amdgcn_wmma_f32_16x16x4_f32(
                false, af, false, bfrag[kc], (short)0, acc, false, false);
        }
        // Epilogue: z-score + segment max. C/D layout: vgpr i holds
        // M = i + 8*(lane>>4), N = lane&15.
        int mbase = r * 16 + 8 * (lane >> 4);
        #pragma unroll
        for (int i = 0; i < 8; ++i) {
            int row = mbase + i;
            float z = (acc[i] - medl[row]) * scl[row];
            atomicMax(&smax[segl[row] * 128 + wave * 16 + (lane & 15)], ordf(z));
        }
    }
    __syncthreads();

    // Pack descending-order sort keys: ties resolved by ascending column.
    for (int t = tid; t < NSEG * 128; t += 256) {
        int s = t >> 7, c = t & 127;
        int col = blockIdx.x * 128 + c;
        unsigned du = ~smax[t];   // ascending du == descending value
        keys[(size_t)s * NMOLS + col] =
            ((unsigned long long)du << 32) | (unsigned)col;
    }
}

// ---------------------------------------------------------------------------
// Kernel 3: one bitonic compare-exchange step over [NSEG][NMOLS] u64 keys.
// ---------------------------------------------------------------------------
__global__ __launch_bounds__(256) void vs_bitonic_step(
    unsigned long long* __restrict__ keys, int j, int k) {
    int i = blockIdx.x * blockDim.x + threadIdx.x;   // 0 .. NSEG*NMOLS-1
    int row = i >> 18;                               // NMOLS = 2^18
    int e = i & (NMOLS - 1);
    int p = e ^ j;
    if (p > e) {
        size_t base = (size_t)row * NMOLS;
        unsigned long long a = keys[base + e];
        unsigned long long b = keys[base + p];
        bool asc = ((e & k) == 0);
        if ((a > b) == asc) { keys[base + e] = b; keys[base + p] = a; }
    }
}

// ---------------------------------------------------------------------------
// Kernel 4: emit top-K values (f32) and indices (i32 bit-pattern) to d_out.
// ---------------------------------------------------------------------------
__global__ __launch_bounds__(256) void vs_writeout(
    const unsigned long long* __restrict__ keys, float* __restrict__ out) {
    int i = blockIdx.x * blockDim.x + threadIdx.x;
    if (i >= NSEG * TOPK) return;
    int s = i / TOPK, e = i - s * TOPK;
    unsigned long long kv = keys[(size_t)s * NMOLS + e];
    unsigned du = (unsigned)(kv >> 32);
    out[i] = unordf(~du);
    ((int*)out)[NSEG * TOPK + i] = (int)(kv & 0xFFFFFFFFu);
}

// ---------------------------------------------------------------------------
extern "C" void kernel_launch(void* const* d_in, const int* in_sizes, int n_in,
                              void* d_out, int out_size, void* d_ws, size_t ws_size,
                              hipStream_t stream) {
    const float* input = (const float*)d_in[0];   // [512,128]
    const float* mols  = (const float*)d_in[1];   // [128,262144]
    const float* rnd   = (const float*)d_in[2];   // [128,10000]
    const int*   pidx  = (const int*)  d_in[3];   // [512]
    (void)in_sizes; (void)n_in; (void)out_size; (void)ws_size;

    // Workspace: u64 keys [32][262144] (64 MB) + medians + scales.
    unsigned long long* keys = (unsigned long long*)d_ws;
    size_t keysBytes = (size_t)NSEG * NMOLS * sizeof(unsigned long long);
    float* medw = (float*)((char*)d_ws + keysBytes);
    float* sclw = medw + PP;

    vs_stats_kernel<<<PP, 256, 0, stream>>>(input, rnd, medw, sclw);

    (void)hipFuncSetAttribute((const void*)vs_fused_gemm_segmax,
                              hipFuncAttributeMaxDynamicSharedMemorySize,
                              K2_SMEM);
    vs_fused_gemm_segmax<<<NMOLS / 128, 256, K2_SMEM, stream>>>(
        input, mols, medw, sclw, pidx, keys);

    // Per-row bitonic sort, 171 steps (NMOLS = 2^18).
    int nblk = (NSEG * NMOLS) / 256;
    for (int k = 2; k <= NMOLS; k <<= 1)
        for (int j = k >> 1; j > 0; j >>= 1)
            vs_bitonic_step<<<nblk, 256, 0, stream>>>(keys, j, k);

    vs_writeout<<<(NSEG * TOPK + 255) / 256, 256, 0, stream>>>(
        keys, (float*)d_out);
}